// InteractionHead_10204842295894
// MI455X (gfx1250) — compile-verified
//
#include <hip/hip_runtime.h>
#include <math.h>

typedef __attribute__((ext_vector_type(2))) float v2f;
typedef __attribute__((ext_vector_type(8))) float v8f;

#define EPSF 1e-10f

__device__ __forceinline__ v8f wmma4(v2f a, v2f b, v8f c) {
  // V_WMMA_F32_16X16X4_F32 : D = A(16x4) * B(4x16) + C(16x16), all fp32
  return __builtin_amdgcn_wmma_f32_16x16x4_f32(false, a, false, b, (short)0, c,
                                               false, false);
}

// Async global->LDS copy (ASYNCcnt path). LDS flat addresses carry the
// wave-relative LDS offset in the low 32 bits (CDNA5 aperture rule), so
// truncating the generic pointer yields the VDST LDS-address operand.
__device__ __forceinline__ void async_ld_b128(const float* lds_dst,
                                              const float* gsrc) {
  unsigned off = (unsigned)(unsigned long long)lds_dst;
  asm volatile("global_load_async_to_lds_b128 %0, %1, off"
               :
               : "v"(off), "v"(gsrc)
               : "memory");
}
__device__ __forceinline__ void wait_async0() {
  asm volatile("s_wait_asynccnt 0" ::: "memory");
}

// ---------------------------------------------------------------------------
// Generic strided-batched GEMM: C = act(A @ B + bias)
// A: MxK (row-major, lda), B: KxN (row-major, ldb), C: MxN (ldc)
// Block = 4 waves = 64 rows x 64 cols; K staged through LDS in panels of 32
// via async b128 copies. Requires M % 64 == 0, N % 64 == 0, K % 4 == 0.
// grid = (N/64, M/64, batch)
// ---------------------------------------------------------------------------
#define KPAN 32
#define ASTR 36  // A LDS row stride (floats): 144B rows, 16B aligned, bank-spread
#define BSTR 68  // B LDS row stride (floats): 272B rows, 16B aligned

__global__ void __launch_bounds__(128)
k_gemm(const float* __restrict__ A, int lda, long sA,
       const float* __restrict__ B, int ldb, long sB,
       const float* __restrict__ bias, int sBias,
       float* __restrict__ C, int ldc, long sC,
       int M, int N, int K, int relu) {
  __shared__ float As[64 * ASTR];
  __shared__ float Bs[KPAN * BSTR];

  const int batch = blockIdx.z;
  A += (long)batch * sA;
  B += (long)batch * sB;
  C += (long)batch * sC;
  if (bias) bias += (long)batch * sBias;

  const int tid = threadIdx.x;
  const int wave = tid >> 5;
  const int lane = tid & 31;
  const int l15 = lane & 15;
  const int hi = lane >> 4;
  const int khalf = 2 * hi;

  const int m0b = blockIdx.y * 64;
  const int n0b = blockIdx.x * 64;
  const int m0 = m0b + wave * 16;

  v8f acc[4];
#pragma unroll
  for (int t = 0; t < 4; ++t) acc[t] = (v8f){0, 0, 0, 0, 0, 0, 0, 0};

  for (int k0 = 0; k0 < K; k0 += KPAN) {
    const int kb = (K - k0 < KPAN) ? (K - k0) : KPAN;  // multiple of 4
    __syncthreads();
    // stage A panel: 64 x kb, k-contiguous rows
    const int cpr = kb >> 2;  // b128 chunks per row
    for (int c = tid; c < 64 * cpr; c += 128) {
      const int m = c / cpr;
      const int kc = (c - m * cpr) * 4;
      async_ld_b128(&As[m * ASTR + kc], A + (long)(m0b + m) * lda + k0 + kc);
    }
    // stage B panel: kb x 64
    for (int c = tid; c < kb * 16; c += 128) {
      const int kr = c >> 4;
      const int nc = (c & 15) * 4;
      async_ld_b128(&Bs[kr * BSTR + nc], B + (long)(k0 + kr) * ldb + n0b + nc);
    }
    wait_async0();
    __syncthreads();

    const float* arow = &As[(wave * 16 + l15) * ASTR];
    for (int kd = 0; kd < kb; kd += 4) {
      const int ka = kd + khalf;
      const v2f a = *(const v2f*)&arow[ka];
      const float* b0 = &Bs[ka * BSTR + l15];
      const float* b1 = b0 + BSTR;
      v2f b;
      b.x = b0[0];  b.y = b1[0];  acc[0] = wmma4(a, b, acc[0]);
      b.x = b0[16]; b.y = b1[16]; acc[1] = wmma4(a, b, acc[1]);
      b.x = b0[32]; b.y = b1[32]; acc[2] = wmma4(a, b, acc[2]);
      b.x = b0[48]; b.y = b1[48]; acc[3] = wmma4(a, b, acc[3]);
    }
  }

#pragma unroll
  for (int t = 0; t < 4; ++t) {
    const int n = n0b + t * 16 + l15;
    const float bv = bias ? bias[n] : 0.0f;
#pragma unroll
    for (int r = 0; r < 8; ++r) {
      const int row = m0 + r + 8 * hi;
      float v = acc[t][r] + bv;
      if (relu) v = fmaxf(v, 0.0f);
      C[(long)row * ldc + n] = v;
    }
  }
}

// ---------------------------------------------------------------------------
// Fused flash attention, fp32 WMMA. One wave per (head, 16-row tile).
// qkv: (2048, 3072) = [Q | K | V], dh = 128, heads = 8, M = N = 2048
// out: (2048, 1024) in (m, h*128 + d) layout. K/V tiles staged via async LDS.
// ---------------------------------------------------------------------------
#define DSTR 132  // 16x128 tile row stride (floats): 528B, 16B aligned

__global__ void __launch_bounds__(32)
k_attn(const float* __restrict__ qkv, float* __restrict__ out) {
  __shared__ float Ks[16 * DSTR];
  __shared__ float Vs[16 * DSTR];
  __shared__ float psh[256];

  const int lane = threadIdx.x;
  const int h = blockIdx.y;
  const int m0 = blockIdx.x * 16;
  const int half = lane >> 4;
  const int l15 = lane & 15;
  const int khalf = 2 * half;
  const float scale = 0.08838834764831845f;  // 1/sqrt(128)

  v2f qa[32];
  {
    const float* qrow = qkv + (long)(m0 + l15) * 3072 + h * 128;
#pragma unroll
    for (int kk = 0; kk < 32; ++kk) {
      const int kd = kk * 4 + khalf;
      qa[kk].x = qrow[kd] * scale;
      qa[kk].y = qrow[kd + 1] * scale;
    }
  }

  v8f acc[8];
#pragma unroll
  for (int t = 0; t < 8; ++t) acc[t] = (v8f){0, 0, 0, 0, 0, 0, 0, 0};
  float mrun[8], lrun[8];
#pragma unroll
  for (int r = 0; r < 8; ++r) { mrun[r] = -1e30f; lrun[r] = 0.0f; }

  for (int j0 = 0; j0 < 2048; j0 += 16) {
    // async-stage K and V tiles (16 x 128) for this column block
#pragma unroll
    for (int c = 0; c < 16; ++c) {
      const int idx = c * 32 + lane;         // 512 b128 chunks per tile, 1024 over 2 tiles
      const int jr = idx >> 5;               // 0..15
      const int dc = (idx & 31) * 4;         // 0..124
      const float* grow = qkv + (long)(j0 + jr) * 3072 + h * 128 + dc;
      async_ld_b128(&Ks[jr * DSTR + dc], grow + 1024);
      async_ld_b128(&Vs[jr * DSTR + dc], grow + 2048);
    }
    wait_async0();  // single wave: its own writes are now visible in LDS

    // S tile = Q(16x128) @ K^T(128x16); K^T fragment = one ds_load_b64
    v8f s = (v8f){0, 0, 0, 0, 0, 0, 0, 0};
    const float* krow = &Ks[l15 * DSTR];
#pragma unroll 8
    for (int kk = 0; kk < 32; ++kk) {
      const int kd = kk * 4 + khalf;
      const v2f b = *(const v2f*)&krow[kd];
      s = wmma4(qa[kk], b, s);
    }
    // online softmax; row r+8*half lives on this lane's 16-lane half
    float p[8];
#pragma unroll
    for (int r = 0; r < 8; ++r) {
      float mx = s[r];
      mx = fmaxf(mx, __shfl_xor(mx, 1));
      mx = fmaxf(mx, __shfl_xor(mx, 2));
      mx = fmaxf(mx, __shfl_xor(mx, 4));
      mx = fmaxf(mx, __shfl_xor(mx, 8));
      const float mnew = fmaxf(mrun[r], mx);
      const float sc = expf(mrun[r] - mnew);
      const float pv = expf(s[r] - mnew);
      float ps = pv;
      ps += __shfl_xor(ps, 1);
      ps += __shfl_xor(ps, 2);
      ps += __shfl_xor(ps, 4);
      ps += __shfl_xor(ps, 8);
      lrun[r] = lrun[r] * sc + ps;
      mrun[r] = mnew;
      p[r] = pv;
#pragma unroll
      for (int t = 0; t < 8; ++t) acc[t][r] *= sc;
    }
    // stage P (C-layout) -> LDS (row-major) -> A-fragments
#pragma unroll
    for (int r = 0; r < 8; ++r) psh[(r + 8 * half) * 16 + l15] = p[r];
    __syncthreads();
    // O += P(16x16) @ V(16x128), V fed from LDS
#pragma unroll
    for (int kk2 = 0; kk2 < 4; ++kk2) {
      const int kd = kk2 * 4 + khalf;
      const v2f a = *(const v2f*)&psh[l15 * 16 + kd];
      const float* v0 = &Vs[kd * DSTR + l15];
      const float* v1 = v0 + DSTR;
#pragma unroll
      for (int t = 0; t < 8; ++t) {
        v2f b;
        b.x = v0[t * 16];
        b.y = v1[t * 16];
        acc[t] = wmma4(a, b, acc[t]);
      }
    }
    __syncthreads();
  }

#pragma unroll
  for (int t = 0; t < 8; ++t)
#pragma unroll
    for (int r = 0; r < 8; ++r) {
      const int row = m0 + r + 8 * half;
      const int col = h * 128 + t * 16 + l15;
      out[(long)row * 1024 + col] = acc[t][r] / lrun[r];
    }
}

// ---------------------------------------------------------------------------
// spatial36: per pair (i,j) of 128 boxes -> 36 features
// ---------------------------------------------------------------------------
__global__ void k_spatial36(const float* __restrict__ boxes,
                            float* __restrict__ sp36) {
  const int p = blockIdx.x * blockDim.x + threadIdx.x;
  if (p >= 128 * 128) return;
  const int i = p >> 7, j = p & 127;
  const float b1x1 = boxes[i * 4 + 0], b1y1 = boxes[i * 4 + 1];
  const float b1x2 = boxes[i * 4 + 2], b1y2 = boxes[i * 4 + 3];
  const float b2x1 = boxes[j * 4 + 0], b2y1 = boxes[j * 4 + 1];
  const float b2x2 = boxes[j * 4 + 2], b2y2 = boxes[j * 4 + 3];
  const float c1x = (b1x1 + b1x2) * 0.5f, c1y = (b1y1 + b1y2) * 0.5f;
  const float c2x = (b2x1 + b2x2) * 0.5f, c2y = (b2y1 + b2y2) * 0.5f;
  const float w1 = b1x2 - b1x1, h1 = b1y2 - b1y1;
  const float w2 = b2x2 - b2x1, h2 = b2y2 - b2y1;
  const float dx = fabsf(c2x - c1x) / (w1 + EPSF);
  const float dy = fabsf(c2y - c1y) / (h1 + EPSF);
  const float whx = fmaxf(fminf(b1x2, b2x2) - fmaxf(b1x1, b2x1), 0.0f);
  const float why = fmaxf(fminf(b1y2, b2y2) - fmaxf(b1y1, b2y1), 0.0f);
  const float inter = whx * why;
  const float a1 = w1 * h1, a2 = w2 * h2;
  const float iou = inter / (a1 + a2 - inter + EPSF);
  float f[18];
  f[0] = c1x / 512.0f; f[1] = c1y / 512.0f;
  f[2] = c2x / 512.0f; f[3] = c2y / 512.0f;
  f[4] = w1 / 512.0f;  f[5] = h1 / 512.0f;
  f[6] = w2 / 512.0f;  f[7] = h2 / 512.0f;
  f[8] = a1 / 262144.0f; f[9] = a2 / 262144.0f;
  f[10] = a2 / (a1 + EPSF);
  f[11] = w1 / (h1 + EPSF);
  f[12] = w2 / (h2 + EPSF);
  f[13] = iou;
  f[14] = (c2x > c1x) ? dx : 0.0f;
  f[15] = (c2x < c1x) ? dx : 0.0f;
  f[16] = (c2y > c1y) ? dy : 0.0f;
  f[17] = (c2y < c1y) ? dy : 0.0f;
#pragma unroll
  for (int t = 0; t < 18; ++t) {
    sp36[(long)p * 36 + t] = f[t];
    sp36[(long)p * 36 + 18 + t] = logf(f[t] + EPSF);
  }
}

// su[h,i] = <u[i,h*64:], aW[h,0:64]>, sj[h,i] = <u[i,h*64:], aW[h,64:128]>
__global__ void k_susj(const float* __restrict__ u, const float* __restrict__ aW,
                       float* __restrict__ su, float* __restrict__ sj) {
  const int idx = blockIdx.x * blockDim.x + threadIdx.x;
  if (idx >= 1024) return;
  const int h = idx >> 7, i = idx & 127;
  float s1 = 0.0f, s2 = 0.0f;
  for (int d = 0; d < 64; ++d) {
    const float uv = u[(long)i * 512 + h * 64 + d];
    s1 += uv * aW[h * 192 + d];
    s2 += uv * aW[h * 192 + 64 + d];
  }
  su[idx] = s1;
  sj[idx] = s2;
}

// scores[h,i,j] = su[h,i] + sj[h,j] + <pw[i*128+j, h*64:], aW[h,128:192]> + ab[h]
__global__ void k_scores(const float* __restrict__ pw, const float* __restrict__ su,
                         const float* __restrict__ sj, const float* __restrict__ aW,
                         const float* __restrict__ ab, float* __restrict__ sc) {
  const int idx = blockIdx.x * blockDim.x + threadIdx.x;
  if (idx >= 131072) return;
  const int h = idx >> 14, i = (idx >> 7) & 127, j = idx & 127;
  float s = su[h * 128 + i] + sj[h * 128 + j] + ab[h];
  const float* pr = pw + ((long)i * 128 + j) * 512 + h * 64;
  const float* ar = aW + h * 192 + 128;
  for (int d = 0; d < 64; ++d) s += pr[d] * ar[d];
  sc[idx] = s;
}

// softmax over i (axis 1) of scores[h,i,j]; one wave per (h,j)
__global__ void __launch_bounds__(32)
k_softmax_col(const float* __restrict__ s, float* __restrict__ w) {
  const int col = blockIdx.x;  // h*128 + j
  const int h = col >> 7, j = col & 127;
  const int lane = threadIdx.x;
  const float* base = s + (long)h * 16384 + j;
  float vals[4];
  float mx = -1e30f;
#pragma unroll
  for (int t = 0; t < 4; ++t) {
    vals[t] = base[(long)(lane + t * 32) * 128];
    mx = fmaxf(mx, vals[t]);
  }
  for (int off = 16; off > 0; off >>= 1) mx = fmaxf(mx, __shfl_xor(mx, off));
  float sum = 0.0f;
#pragma unroll
  for (int t = 0; t < 4; ++t) { vals[t] = expf(vals[t] - mx); sum += vals[t]; }
  for (int off = 16; off > 0; off >>= 1) sum += __shfl_xor(sum, off);
  const float inv = 1.0f / sum;
  float* wb = w + (long)h * 16384 + j;
#pragma unroll
  for (int t = 0; t < 4; ++t) wb[(long)(lane + t * 32) * 128] = vals[t] * inv;
}

// t[h,j,d] = sum_i wts[h,i,j] * u[i,h*64+d] * pw[i*128+j, h*64+d]
__global__ void k_wsum_t(const float* __restrict__ wts, const float* __restrict__ u,
                         const float* __restrict__ pw, float* __restrict__ t) {
  const int idx = blockIdx.x * blockDim.x + threadIdx.x;
  if (idx >= 65536) return;
  const int h = idx >> 13, j = (idx >> 6) & 127, d = idx & 63;
  float acc = 0.0f;
  for (int i = 0; i < 128; ++i) {
    acc += wts[(long)h * 16384 + i * 128 + j] *
           u[(long)i * 512 + h * 64 + d] *
           pw[((long)i * 128 + j) * 512 + h * 64 + d];
  }
  t[idx] = acc;
}

// out = LayerNorm(x + h) * g + b, one wave per row
__global__ void __launch_bounds__(32)
k_add_ln(const float* __restrict__ x, const float* __restrict__ h,
         const float* __restrict__ g, const float* __restrict__ b,
         float* __restrict__ out, int cols) {
  const int row = blockIdx.x;
  const int lane = threadIdx.x;
  const float* xr = x + (long)row * cols;
  const float* hr = h + (long)row * cols;
  float sum = 0.0f;
  for (int c = lane; c < cols; c += 32) sum += xr[c] + hr[c];
  for (int off = 16; off > 0; off >>= 1) sum += __shfl_xor(sum, off);
  const float mean = sum / (float)cols;
  float vs = 0.0f;
  for (int c = lane; c < cols; c += 32) {
    const float d = xr[c] + hr[c] - mean;
    vs += d * d;
  }
  for (int off = 16; off > 0; off >>= 1) vs += __shfl_xor(vs, off);
  const float rstd = rsqrtf(vs / (float)cols + 1e-5f);
  for (int c = lane; c < cols; c += 32)
    out[(long)row * cols + c] = (xr[c] + hr[c] - mean) * rstd * g[c] + b[c];
}

// appin[m] = [x[px[m]], x[py[m]]]  (2048 x 1024)
__global__ void k_gather_pairs(const float* __restrict__ x,
                               const int* __restrict__ rel,
                               float* __restrict__ appin) {
  const int idx = blockIdx.x * blockDim.x + threadIdx.x;
  if (idx >= 2048 * 1024) return;
  const int m = idx >> 10, c = idx & 1023;
  const int px = rel[m * 2 + 0], py = rel[m * 2 + 1];
  appin[idx] = (c < 512) ? x[(long)px * 512 + c] : x[(long)py * 512 + (c - 512)];
}

// sppair[m] = y[px[m]*128 + py[m]]  (2048 x 512)
__global__ void k_gather_sppair(const float* __restrict__ y,
                                const int* __restrict__ rel,
                                float* __restrict__ sp) {
  const int idx = blockIdx.x * blockDim.x + threadIdx.x;
  if (idx >= 2048 * 512) return;
  const int m = idx >> 9, c = idx & 511;
  const int px = rel[m * 2 + 0], py = rel[m * 2 + 1];
  sp[idx] = y[((long)px * 128 + py) * 512 + c];
}

// (16, Kdim, 32) -> (Kdim, 512) with col = k*32+s
__global__ void k_permute_kbs(const float* __restrict__ src,
                              float* __restrict__ dst, int Kdim) {
  const long idx = (long)blockIdx.x * blockDim.x + threadIdx.x;
  if (idx >= (long)Kdim * 512) return;
  const long bcol = idx >> 9;
  const int c = (int)(idx & 511);
  const int k = c >> 5, sidx = c & 31;
  dst[idx] = src[(long)k * Kdim * 32 + bcol * 32 + sidx];
}

// out[c] = sum_k f3b[k*512 + c]
__global__ void k_sumf3b(const float* __restrict__ f3b, float* __restrict__ out) {
  const int c = blockIdx.x * blockDim.x + threadIdx.x;
  if (c >= 512) return;
  float acc = 0.0f;
  for (int k = 0; k < 16; ++k) acc += f3b[k * 512 + c];
  out[c] = acc;
}

// a1g[c] = sum_d g[d] * f1Wg[k(c), d, s(c)] + f1bg[c]
__global__ void k_f1g(const float* __restrict__ g, const float* __restrict__ f1Wg,
                      const float* __restrict__ f1bg, float* __restrict__ a1g) {
  const int c = blockIdx.x * blockDim.x + threadIdx.x;
  if (c >= 512) return;
  const int k = c >> 5, sidx = c & 31;
  float acc = f1bg[c];
  const float* w = f1Wg + (long)k * 1280 * 32 + sidx;
  for (int d = 0; d < 1280; ++d) acc += g[d] * w[(long)d * 32];
  a1g[c] = acc;
}

__global__ void k_mul_relu(const float* __restrict__ a, const float* __restrict__ b,
                           float* __restrict__ o, int total) {
  const int idx = blockIdx.x * blockDim.x + threadIdx.x;
  if (idx >= total) return;
  o[idx] = fmaxf(a[idx] * b[idx], 0.0f);
}

__global__ void k_mulrow_relu(const float* __restrict__ a,
                              const float* __restrict__ rowv,
                              float* __restrict__ o, int total) {
  const int idx = blockIdx.x * blockDim.x + threadIdx.x;
  if (idx >= total) return;
  o[idx] = fmaxf(a[idx] * rowv[idx & 511], 0.0f);
}

// ---------------------------------------------------------------------------
static inline void gemm(hipStream_t st, const float* A, int lda, long sA,
                        const float* B, int ldb, long sB,
                        const float* bias, int sBias,
                        float* C, int ldc, long sC,
                        int M, int N, int K, int relu, int batch) {
  dim3 g(N / 64, M / 64, batch), b(128, 1, 1);
  k_gemm<<<g, b, 0, st>>>(A, lda, sA, B, ldb, sB, bias, sBias, C, ldc, sC,
                          M, N, K, relu);
}

extern "C" void kernel_launch(void* const* d_in, const int* in_sizes, int n_in,
                              void* d_out, int out_size, void* d_ws, size_t ws_size,
                              hipStream_t stream) {
  (void)in_sizes; (void)n_in; (void)out_size; (void)ws_size;
  const float* obj_ctx = (const float*)d_in[0];
  const float* boxes   = (const float*)d_in[1];
  const float* gfeat   = (const float*)d_in[2];
  const float* sp_W1 = (const float*)d_in[3];  const float* sp_b1 = (const float*)d_in[4];
  const float* sp_W2 = (const float*)d_in[5];  const float* sp_b2 = (const float*)d_in[6];
  const float* sp_W3 = (const float*)d_in[7];  const float* sp_b3 = (const float*)d_in[8];
  const float* c_uW  = (const float*)d_in[9];  const float* c_ub  = (const float*)d_in[10];
  const float* c_pW  = (const float*)d_in[11]; const float* c_pb  = (const float*)d_in[12];
  const float* c_aW  = (const float*)d_in[13]; const float* c_ab  = (const float*)d_in[14];
  const float* c_mW  = (const float*)d_in[15]; const float* c_mb  = (const float*)d_in[16];
  const float* c_gW  = (const float*)d_in[17]; const float* c_gb  = (const float*)d_in[18];
  const float* c_l1g = (const float*)d_in[19]; const float* c_l1b = (const float*)d_in[20];
  const float* c_fW1 = (const float*)d_in[21]; const float* c_fb1 = (const float*)d_in[22];
  const float* c_fW2 = (const float*)d_in[23]; const float* c_fb2 = (const float*)d_in[24];
  const float* c_l2g = (const float*)d_in[25]; const float* c_l2b = (const float*)d_in[26];
  const float* m1W = (const float*)d_in[27];  const float* m1b = (const float*)d_in[28];
  const float* m2W = (const float*)d_in[29];  const float* m2b = (const float*)d_in[30];
  const float* m3W = (const float*)d_in[31];  const float* m3b = (const float*)d_in[32];
  const float* g1W = (const float*)d_in[33];  const float* g1b = (const float*)d_in[34];
  const float* g2W = (const float*)d_in[35];  const float* g2b = (const float*)d_in[36];
  const float* g3W = (const float*)d_in[37];  const float* g3b = (const float*)d_in[38];
  const float* rc_W = (const float*)d_in[39]; const float* rc_b = (const float*)d_in[40];
  const float* Wqkv = (const float*)d_in[41]; const float* bqkv = (const float*)d_in[42];
  const float* Wo   = (const float*)d_in[43]; const float* bo   = (const float*)d_in[44];
  const float* p_l1g = (const float*)d_in[45]; const float* p_l1b = (const float*)d_in[46];
  const float* pfW1 = (const float*)d_in[47]; const float* pfb1 = (const float*)d_in[48];
  const float* pfW2 = (const float*)d_in[49]; const float* pfb2 = (const float*)d_in[50];
  const float* p_l2g = (const float*)d_in[51]; const float* p_l2b = (const float*)d_in[52];
  const int* rel = (const int*)d_in[53];
  float* out = (float*)d_out;

  // ---- workspace bump allocator (floats) ----
  float* ws = (float*)d_ws;
  long off = 0;
  auto alloc = [&](long n) { float* p = ws + off; off += n; return p; };
  float* y     = alloc(16384L * 512);   // persistent (both layers + sp_pair)
  float* pwbuf = alloc(16384L * 512);   // per-layer pair act; later aliased as qkv
  float* big3  = alloc(6881280);        // sp36|h1|h2 ; later attn_out|ffh2
  float* xA = alloc(65536); float* xB = alloc(65536);
  float* u  = alloc(65536);
  float* su = alloc(1024);  float* sj = alloc(1024);
  float* sc = alloc(131072); float* wts = alloc(131072);
  float* tvec = alloc(65536);
  float* agg2 = alloc(65536); float* agg3 = alloc(65536);
  float* ffh  = alloc(131072);
  float* appin = alloc(2097152);        // later aliased as proj
  float* app   = alloc(1048576);
  float* sppair = alloc(1048576);
  float* A1 = alloc(1048576); float* A2 = alloc(1048576); float* hm = alloc(1048576);
  float* tok = alloc(2097152);
  float* F1t = alloc(262144); float* F2t = alloc(262144); float* F2gt = alloc(262144);
  float* a1g = alloc(512); float* biasO = alloc(512); float* biasOg = alloc(512);
  float* tbuf = alloc(2097152);

  float* sp36 = big3;
  float* hh1 = big3 + 589824;
  float* hh2 = big3 + 589824 + 2097152;
  float* qkv = pwbuf;            // 2048*3072 <= 16384*512
  float* attn_out = big3;        // 2048*1024
  float* ffh2 = big3 + 2097152;  // 2048*2048
  float* proj = appin;           // 2048*1024

  // ---- spatial MLP: sp36 -> 128 -> 256 -> 512 (=> y) ----
  k_spatial36<<<64, 256, 0, stream>>>(boxes, sp36);
  gemm(stream, sp36, 36, 0, sp_W1, 128, 0, sp_b1, 0, hh1, 128, 0, 16384, 128, 36, 1, 1);
  gemm(stream, hh1, 128, 0, sp_W2, 256, 0, sp_b2, 0, hh2, 256, 0, 16384, 256, 128, 1, 1);
  gemm(stream, hh2, 256, 0, sp_W3, 512, 0, sp_b3, 0, y, 512, 0, 16384, 512, 256, 1, 1);

  // ---- coop encoder (2 layers) ----
  const float* xin = obj_ctx;
  for (int l = 0; l < 2; ++l) {
    const float* uW = c_uW + (long)l * 262144; const float* ub = c_ub + l * 512;
    const float* pW = c_pW + (long)l * 262144; const float* pb = c_pb + l * 512;
    const float* aW = c_aW + (long)l * 1536;   const float* ab = c_ab + l * 8;
    const float* mW = c_mW + (long)l * 32768;  const float* mb = c_mb + l * 512;
    const float* gW = c_gW + (long)l * 262144; const float* gb = c_gb + l * 512;
    const float* fW1 = c_fW1 + (long)l * 524288; const float* fb1 = c_fb1 + l * 1024;
    const float* fW2 = c_fW2 + (long)l * 524288; const float* fb2 = c_fb2 + l * 512;

    gemm(stream, xin, 512, 0, uW, 512, 0, ub, 0, u, 512, 0, 128, 512, 512, 1, 1);
    gemm(stream, y, 512, 0, pW, 512, 0, pb, 0, pwbuf, 512, 0, 16384, 512, 512, 1, 1);
    k_susj<<<4, 256, 0, stream>>>(u, aW, su, sj);
    k_scores<<<512, 256, 0, stream>>>(pwbuf, su, sj, aW, ab, sc);
    k_softmax_col<<<1024, 32, 0, stream>>>(sc, wts);
    k_wsum_t<<<256, 256, 0, stream>>>(wts, u, pwbuf, tvec);
    // agg2[j, h*64+e] = relu(sum_d t[h,j,d] * mW[h,d,e] + mb[h,e])  (batched over h)
    gemm(stream, tvec, 64, 8192, mW, 64, 4096, mb, 64, agg2, 512, 64, 128, 64, 64, 1, 8);
    gemm(stream, agg2, 512, 0, gW, 512, 0, gb, 0, agg3, 512, 0, 128, 512, 512, 0, 1);
    k_add_ln<<<128, 32, 0, stream>>>(xin, agg3, c_l1g + l * 512, c_l1b + l * 512, xB, 512);
    gemm(stream, xB, 512, 0, fW1, 1024, 0, fb1, 0, ffh, 1024, 0, 128, 1024, 512, 1, 1);
    gemm(stream, ffh, 1024, 0, fW2, 512, 0, fb2, 0, agg3, 512, 0, 128, 512, 1024, 0, 1);
    k_add_ln<<<128, 32, 0, stream>>>(xB, agg3, c_l2g + l * 512, c_l2b + l * 512, xA, 512);
    xin = xA;
  }

  // ---- pair features ----
  k_gather_pairs<<<8192, 256, 0, stream>>>(xA, rel, appin);
  gemm(stream, appin, 1024, 0, rc_W, 512, 0, rc_b, 0, app, 512, 0, 2048, 512, 1024, 0, 1);
  k_gather_sppair<<<4096, 256, 0, stream>>>(y, rel, sppair);

  // ---- mbf (app x sp_pair) -> tok[:, :512] ----
  k_permute_kbs<<<1024, 256, 0, stream>>>(m1W, F1t, 512);
  k_permute_kbs<<<1024, 256, 0, stream>>>(m2W, F2t, 512);
  k_permute_kbs<<<1024, 256, 0, stream>>>(g2W, F2gt, 512);
  k_sumf3b<<<2, 256, 0, stream>>>(m3b, biasO);
  k_sumf3b<<<2, 256, 0, stream>>>(g3b, biasOg);
  k_f1g<<<2, 256, 0, stream>>>(gfeat, g1W, g1b, a1g);

  gemm(stream, app, 512, 0, F1t, 512, 0, m1b, 0, A1, 512, 0, 2048, 512, 512, 0, 1);
  gemm(stream, sppair, 512, 0, F2t, 512, 0, m2b, 0, A2, 512, 0, 2048, 512, 512, 0, 1);
  k_mul_relu<<<4096, 256, 0, stream>>>(A1, A2, hm, 2048 * 512);
  gemm(stream, hm, 512, 0, m3W, 512, 0, biasO, 0, tok, 1024, 0, 2048, 512, 512, 1, 1);

  // ---- mbfg (global x sp_pair) -> tok[:, 512:] ----
  gemm(stream, sppair, 512, 0, F2gt, 512, 0, g2b, 0, A1, 512, 0, 2048, 512, 512, 0, 1);
  k_mulrow_relu<<<4096, 256, 0, stream>>>(A1, a1g, hm, 2048 * 512);
  gemm(stream, hm, 512, 0, g3W, 512, 0, biasOg, 0, tok + 512, 1024, 0, 2048, 512, 512, 1, 1);

  // ---- cmp transformer block ----
  gemm(stream, tok, 1024, 0, Wqkv, 3072, 0, bqkv, 0, qkv, 3072, 0, 2048, 3072, 1024, 0, 1);
  k_attn<<<dim3(128, 8), 32, 0, stream>>>(qkv, attn_out);
  gemm(stream, attn_out, 1024, 0, Wo, 1024, 0, bo, 0, proj, 1024, 0, 2048, 1024, 1024, 0, 1);
  k_add_ln<<<2048, 32, 0, stream>>>(tok, proj, p_l1g, p_l1b, tbuf, 1024);
  gemm(stream, tbuf, 1024, 0, pfW1, 2048, 0, pfb1, 0, ffh2, 2048, 0, 2048, 2048, 1024, 1, 1);
  gemm(stream, ffh2, 2048, 0, pfW2, 1024, 0, pfb2, 0, proj, 1024, 0, 2048, 1024, 2048, 0, 1);
  k_add_ln<<<2048, 32, 0, stream>>>(tbuf, proj, p_l2g, p_l2b, out, 1024);
}